// pose_estimate_loss_39711267619448
// MI455X (gfx1250) — compile-verified
//
#include <hip/hip_runtime.h>
#include <hip/hip_bf16.h>

typedef __attribute__((ext_vector_type(2))) float v2f;
typedef __attribute__((ext_vector_type(8))) float v8f;

#define NBLK 1024
#define NTHR 256

// ---------------------------------------------------------------------------
// Pass 1: per-block partial Huber sums over trilinearly-interpolated SDF.
// Memory-bound: 12B/point coalesced stream + 8 L2-resident gathers/point.
// ---------------------------------------------------------------------------
__global__ __launch_bounds__(NTHR) void sdf_huber_partial(
    const float* __restrict__ vox,      // [128,64,64] C-order
    const float* __restrict__ pts,      // [N,3]
    const int*   __restrict__ hgt,      // scalar height_gt
    float*       __restrict__ partial,  // [NBLK]
    int n)
{
  const float zoff = 0.5f * (float)(*hgt);   // uniform scalar load
  float acc = 0.0f;
  const int stride = gridDim.x * blockDim.x;
  for (int i = blockIdx.x * blockDim.x + threadIdx.x; i < n; i += stride) {
    const float* p = pts + 3 * i;            // contiguous 12B per lane
    float x = p[0] + 6.0f;                   // + LENGTH/2
    float y = p[1] + 3.0f;                   // + WIDTH/2
    float z = p[2] + zoff;                   // + height_gt/2
    float xm = floorf(x * 10.0f);            // floor(x / 0.1)
    float ym = floorf(y * 10.0f);
    float zm = floorf(z * 10.0f);
    float ux = (x - xm * 0.1f) * 10.0f;      // frac weight of 'max' corner
    float uy = (y - ym * 0.1f) * 10.0f;
    float uz = (z - zm * 0.1f) * 10.0f;
    // clamp AFTER frac, as in the reference
    int x0 = (int)fminf(fmaxf(xm,        0.0f), 127.0f);
    int y0 = (int)fminf(fmaxf(ym,        0.0f),  63.0f);
    int z0 = (int)fminf(fmaxf(zm,        0.0f),  63.0f);
    int x1 = (int)fminf(fmaxf(xm + 1.0f, 0.0f), 127.0f);
    int y1 = (int)fminf(fmaxf(ym + 1.0f, 0.0f),  63.0f);
    int z1 = (int)fminf(fmaxf(zm + 1.0f, 0.0f),  63.0f);
    int bx0 = x0 << 12, bx1 = x1 << 12;      // x * 64*64
    int by0 = y0 << 6,  by1 = y1 << 6;       // y * 64
    float c000 = vox[bx0 + by0 + z0];
    float c001 = vox[bx0 + by0 + z1];
    float c010 = vox[bx0 + by1 + z0];
    float c011 = vox[bx0 + by1 + z1];
    float c100 = vox[bx1 + by0 + z0];
    float c101 = vox[bx1 + by0 + z1];
    float c110 = vox[bx1 + by1 + z0];
    float c111 = vox[bx1 + by1 + z1];
    float vx = 1.0f - ux, vy = 1.0f - uy, vz = 1.0f - uz;
    float s =
        c111 * (ux * uy * uz) + c110 * (ux * uy * vz) +
        c101 * (ux * vy * uz) + c100 * (ux * vy * vz) +
        c011 * (vx * uy * uz) + c010 * (vx * uy * vz) +
        c001 * (vx * vy * uz) + c000 * (vx * vy * vz);
    float a = fabsf(s);
    acc += (a < 1.0f) ? (0.5f * s * s) : (a - 0.5f);   // Huber, delta=1
  }

  // wave32 tree reduction
  for (int o = 16; o > 0; o >>= 1) acc += __shfl_down(acc, o, 32);

  __shared__ float smem[NTHR / 32];
  const int lane = threadIdx.x & 31;
  const int wave = threadIdx.x >> 5;
  if (lane == 0) smem[wave] = acc;
  __syncthreads();
  if (threadIdx.x == 0) {
    float s = 0.0f;
#pragma unroll
    for (int w = 0; w < NTHR / 32; ++w) s += smem[w];
    partial[blockIdx.x] = s;
  }
}

// ---------------------------------------------------------------------------
// Pass 2: single-wave final reduction using V_WMMA_F32_16X16X4_F32.
// Layout-agnostic identity: with B == all-ones, sum(D) == 16 * sum(A),
// where A's 64 elems live in (32 lanes x 2 VGPRs) and D's 256 elems in
// (32 lanes x 8 VGPRs). EXEC is all-ones: one full wave, no divergence.
// ---------------------------------------------------------------------------
__global__ __launch_bounds__(32) void final_reduce_wmma(
    const float* __restrict__ partial, float* __restrict__ out, float inv_n)
{
  const int tid = threadIdx.x;  // 0..31
  float a0 = 0.0f, a1 = 0.0f;
#pragma unroll
  for (int j = 0; j < 32; j += 2) {        // 1024 partials -> 64 lane values
    a0 += partial[tid + j * 32];
    a1 += partial[tid + (j + 1) * 32];
  }
  v2f A = {a0, a1};
  v2f B = {1.0f, 1.0f};                    // all-ones 4x16 B, layout-independent
  v8f C = {};
  v8f D = __builtin_amdgcn_wmma_f32_16x16x4_f32(
      /*neg_a=*/false, A, /*neg_b=*/false, B,
      /*c_mod=*/(short)0, C, /*reuse_a=*/false, /*reuse_b=*/false);
  float s = D[0] + D[1] + D[2] + D[3] + D[4] + D[5] + D[6] + D[7];
  for (int o = 16; o > 0; o >>= 1) s += __shfl_down(s, o, 32);
  if (tid == 0) out[0] = s * (1.0f / 16.0f) * inv_n;  // undo x16, take mean
}

// ---------------------------------------------------------------------------
extern "C" void kernel_launch(void* const* d_in, const int* in_sizes, int n_in,
                              void* d_out, int out_size, void* d_ws, size_t ws_size,
                              hipStream_t stream) {
  (void)n_in; (void)out_size; (void)ws_size;
  const float* vox = (const float*)d_in[0];   // voxels [128,64,64] f32
  const float* pts = (const float*)d_in[1];   // pts_centroid [N,3] f32
  const int*   hgt = (const int*)d_in[2];     // height_gt scalar (int)
  const int n = in_sizes[1] / 3;              // number of points
  float* partial = (float*)d_ws;              // NBLK floats of scratch

  sdf_huber_partial<<<NBLK, NTHR, 0, stream>>>(vox, pts, hgt, partial, n);
  final_reduce_wmma<<<1, 32, 0, stream>>>(partial, (float*)d_out,
                                          1.0f / (float)n);
}